// waveletTransformModule_14096082665717
// MI455X (gfx1250) — compile-verified
//
#include <hip/hip_runtime.h>

typedef float v2f __attribute__((ext_vector_type(2)));
typedef float v4f __attribute__((ext_vector_type(4)));
typedef float v8f __attribute__((ext_vector_type(8)));

#define IMGS 96
// DWT output (conv1 input) with 1-px halo: [96][258][258]
#define HA 258
#define CA_STRIDE (HA * HA)          // 66564
// conv1 output (conv2 input) with 1-px halo, 16ch NHWC: [96][130][130][16]
#define HB 130
#define H1_STRIDE (HB * HB * 16)     // 270400
// conv2 output 8ch NHWC: [96][64][64][8]

// workspace float offsets
#define WS_CA   ((size_t)0)
#define WS_H1   ((size_t)6390272)                 // 96*66564 = 6390144, +slack
#define WS_H2   (WS_H1 + (size_t)96 * H1_STRIDE)  // +25958400
#define WS_WB2  (WS_H2 + (size_t)3145728)
#define WS_WB1  (WS_WB2 + (size_t)2304)

// ---------------------------------------------------------------------------
// Kernel 1: Haar DWT. cA -> halo layout in ws; cH/cV -> "high" output (NT).
// ---------------------------------------------------------------------------
__global__ __launch_bounds__(256)
void k_dwt(const float* __restrict__ x, float* __restrict__ cA,
           float* __restrict__ high) {
    int idx = blockIdx.x * blockDim.x + threadIdx.x;   // 96*256*256
    int j   = idx & 255;
    int i   = (idx >> 8) & 255;
    int img = idx >> 16;

    const float* p0 = x + (((size_t)img * 512) + 2 * i) * 512 + 2 * j;
    v2f r0 = __builtin_nontemporal_load((const v2f*)p0);
    v2f r1 = __builtin_nontemporal_load((const v2f*)(p0 + 512));
    float a = r0[0], b = r0[1], c = r1[0], d = r1[1];

    float vA = (a + b + c + d) * 0.5f;
    float vH = (a + b - c - d) * 0.5f;
    float vV = (a - b + c - d) * 0.5f;

    cA[(size_t)img * CA_STRIDE + (size_t)(i + 1) * HA + (j + 1)] = vA;
    size_t hb = (size_t)img * (2 * 65536) + (size_t)i * 256 + j;
    __builtin_nontemporal_store(vH, &high[hb]);
    __builtin_nontemporal_store(vV, &high[hb + 65536]);
}

// ---------------------------------------------------------------------------
// Halo zeroing: cA border (96 * (2*258 + 2*256) cells)
// ---------------------------------------------------------------------------
__global__ __launch_bounds__(256)
void k_zero_ca(float* __restrict__ cA) {
    int idx = blockIdx.x * blockDim.x + threadIdx.x;
    if (idx >= 96 * 1028) return;
    int cell = idx % 1028;
    int img  = idx / 1028;
    int row, col;
    if (cell < 258)      { row = 0;   col = cell; }
    else if (cell < 516) { row = 257; col = cell - 258; }
    else {
        int q = cell - 516;           // 0..511
        row = 1 + (q & 255);
        col = (q < 256) ? 0 : 257;
    }
    cA[(size_t)img * CA_STRIDE + (size_t)row * HA + col] = 0.0f;
}

// ---------------------------------------------------------------------------
// Halo zeroing: h1 border (96 * (2*130 + 2*128) cells * 16 ch) = 792576
// ---------------------------------------------------------------------------
__global__ __launch_bounds__(256)
void k_zero_h1(float* __restrict__ h1) {
    int idx = blockIdx.x * blockDim.x + threadIdx.x;   // exact 3096*256
    int ch   = idx & 15;
    int cell = (idx >> 4) % 516;
    int img  = (idx >> 4) / 516;
    int row, col;
    if (cell < 130)      { row = 0;   col = cell; }
    else if (cell < 260) { row = 129; col = cell - 130; }
    else {
        int q = cell - 260;           // 0..255
        row = 1 + (q & 127);
        col = (q < 128) ? 0 : 129;
    }
    h1[(size_t)img * H1_STRIDE + ((size_t)row * HB + col) * 16 + ch] = 0.0f;
}

// ---------------------------------------------------------------------------
// Pre-pack B fragments (zero-padded, lane-contiguous):
//  wB2[tap][hi][oc16][8] : ic = 4*(o>>1) + (o&1) + 2*hi ; oc>=8 -> 0
//  wB1[ky][hi][oc16][2]  : kx = v + 2*hi ; kx==3 -> 0
// ---------------------------------------------------------------------------
__global__ __launch_bounds__(256)
void k_prepw(const float* __restrict__ w1, const float* __restrict__ w2,
             float* __restrict__ wB1, float* __restrict__ wB2) {
    int idx = blockIdx.x * blockDim.x + threadIdx.x;
    if (idx < 2304) {
        int o   = idx & 7;
        int n   = (idx >> 3) & 15;
        int hi  = (idx >> 7) & 1;
        int tap = idx >> 8;
        int ic  = 4 * (o >> 1) + (o & 1) + 2 * hi;
        int ky  = tap / 3, kx = tap % 3;
        wB2[idx] = (n < 8) ? w2[((n * 16 + ic) * 3 + ky) * 3 + kx] : 0.0f;
    } else if (idx < 2304 + 192) {
        int j  = idx - 2304;
        int v  = j & 1;
        int n  = (j >> 1) & 15;
        int hi = (j >> 5) & 1;
        int ky = j >> 6;
        int kx = v + 2 * hi;
        wB1[j] = (kx < 3) ? w1[n * 9 + ky * 3 + kx] : 0.0f;
    }
}

// ---------------------------------------------------------------------------
// Kernel 2: conv 3x3 (1->16) + bias + ReLU + maxpool2 via WMMA f32 16x16x4.
// K padded 9->12 as k = ky*4 + kx (kx==3 zero pad). Wave: 16 px x 2 rows ->
// 8 pooled px x 16 ch. h1 stored NHWC (permuted channels) with halo.
// ---------------------------------------------------------------------------
__global__ __launch_bounds__(256)
void k_conv1(const float* __restrict__ cA, const float* __restrict__ wB1,
             const float* __restrict__ b1, float* __restrict__ h1) {
    int gtid = blockIdx.x * blockDim.x + threadIdx.x;
    int wave = gtid >> 5;
    int lane = gtid & 31;
    int t   = wave & 15;            // x-tile
    int r   = (wave >> 4) & 127;    // pooled row
    int img = wave >> 11;           // 0..95
    int x0  = t * 16;
    int mlo = lane & 15;            // A: M (pixel), B/D: N (oc)
    int hi  = lane >> 4;

    const float* in = cA + (size_t)img * CA_STRIDE;
    int colbase = x0 + mlo + 2 * hi;   // halo col for v=0

    v8f acc0 = {}; v8f acc1 = {};
    #pragma unroll
    for (int ky = 0; ky < 3; ++ky) {
        v2f bf = *(const v2f*)(wB1 + (((ky * 2 + hi) * 16) + mlo) * 2);
        #pragma unroll
        for (int rr = 0; rr < 2; ++rr) {
            int row = 2 * r + rr + ky;            // halo row (iy+1)
            const float* p = in + (size_t)row * HA + colbase;
            float a0 = p[0];
            float a1 = p[1];
            v2f af;
            af[0] = a0;
            af[1] = hi ? 0.0f : a1;               // kx==3 pad lane-half
            if (rr == 0)
                acc0 = __builtin_amdgcn_wmma_f32_16x16x4_f32(
                    false, af, false, bf, (short)0, acc0, false, false);
            else
                acc1 = __builtin_amdgcn_wmma_f32_16x16x4_f32(
                    false, af, false, bf, (short)0, acc1, false, false);
        }
    }

    // bias + relu + 2x2 maxpool; store permuted-channel NHWC with halo
    float bias = b1[mlo];
    int c3 = mlo & 3;
    int g2 = (mlo >> 2) << 1;
    int sIdx = (c3 < 2) ? (g2 + c3) : (6 + g2 + c3);   // storage slot
    float* outb = h1 + (size_t)img * H1_STRIDE + ((size_t)(r + 1) * HB) * 16;
    #pragma unroll
    for (int p = 0; p < 4; ++p) {
        float v00 = fmaxf(acc0[2 * p]     + bias, 0.0f);
        float v01 = fmaxf(acc0[2 * p + 1] + bias, 0.0f);
        float v10 = fmaxf(acc1[2 * p]     + bias, 0.0f);
        float v11 = fmaxf(acc1[2 * p + 1] + bias, 0.0f);
        float pooled = fmaxf(fmaxf(v00, v01), fmaxf(v10, v11));
        int px = t * 8 + p + 4 * hi;
        outb[(px + 1) * 16 + sIdx] = pooled;
    }
}

// ---------------------------------------------------------------------------
// Kernel 3: conv 3x3 (16->8, N padded to 16) + bias + ReLU + maxpool2.
// Per-tap GEMMs, K=16 (4 k-steps). A: two b128 loads per tap/row thanks to
// channel permutation. B: two b128 loads per tap from pre-packed wB2.
// 72 WMMA per wave. Output h2 NHWC [96][64][64][8].
// ---------------------------------------------------------------------------
__global__ __launch_bounds__(256)
void k_conv2(const float* __restrict__ h1, const float* __restrict__ wB2,
             const float* __restrict__ b2, float* __restrict__ h2) {
    int gtid = blockIdx.x * blockDim.x + threadIdx.x;
    int wave = gtid >> 5;
    int lane = gtid & 31;
    int t   = wave & 7;
    int r   = (wave >> 3) & 63;
    int img = wave >> 9;
    int x0  = t * 16;
    int mlo = lane & 15;
    int hi  = lane >> 4;

    const float* in = h1 + (size_t)img * H1_STRIDE;
    const float* wb = wB2 + hi * 128 + mlo * 8;

    v8f acc0 = {}; v8f acc1 = {};
    #pragma unroll
    for (int tap = 0; tap < 9; ++tap) {
        const int ky = tap / 3, kx = tap % 3;       // compile-time
        const float* wp = wb + tap * 256;
        v4f g0 = *(const v4f*)(wp);
        v4f g1 = *(const v4f*)(wp + 4);
        v2f bf0 = __builtin_shufflevector(g0, g0, 0, 1);
        v2f bf1 = __builtin_shufflevector(g0, g0, 2, 3);
        v2f bf2 = __builtin_shufflevector(g1, g1, 0, 1);
        v2f bf3 = __builtin_shufflevector(g1, g1, 2, 3);
        #pragma unroll
        for (int rr = 0; rr < 2; ++rr) {
            int row = 2 * r + rr + ky;              // halo row
            int col = x0 + mlo + kx;                // halo col
            const float* ap = in + ((size_t)row * HB + col) * 16 + 8 * hi;
            v4f f0 = *(const v4f*)(ap);
            v4f f1 = *(const v4f*)(ap + 4);
            v2f af0 = __builtin_shufflevector(f0, f0, 0, 1);
            v2f af1 = __builtin_shufflevector(f0, f0, 2, 3);
            v2f af2 = __builtin_shufflevector(f1, f1, 0, 1);
            v2f af3 = __builtin_shufflevector(f1, f1, 2, 3);
            if (rr == 0) {
                acc0 = __builtin_amdgcn_wmma_f32_16x16x4_f32(false, af0, false, bf0, (short)0, acc0, false, false);
                acc0 = __builtin_amdgcn_wmma_f32_16x16x4_f32(false, af1, false, bf1, (short)0, acc0, false, false);
                acc0 = __builtin_amdgcn_wmma_f32_16x16x4_f32(false, af2, false, bf2, (short)0, acc0, false, false);
                acc0 = __builtin_amdgcn_wmma_f32_16x16x4_f32(false, af3, false, bf3, (short)0, acc0, false, false);
            } else {
                acc1 = __builtin_amdgcn_wmma_f32_16x16x4_f32(false, af0, false, bf0, (short)0, acc1, false, false);
                acc1 = __builtin_amdgcn_wmma_f32_16x16x4_f32(false, af1, false, bf1, (short)0, acc1, false, false);
                acc1 = __builtin_amdgcn_wmma_f32_16x16x4_f32(false, af2, false, bf2, (short)0, acc1, false, false);
                acc1 = __builtin_amdgcn_wmma_f32_16x16x4_f32(false, af3, false, bf3, (short)0, acc1, false, false);
            }
        }
    }

    float bias = b2[mlo & 7];
    #pragma unroll
    for (int p = 0; p < 4; ++p) {
        float v00 = fmaxf(acc0[2 * p]     + bias, 0.0f);
        float v01 = fmaxf(acc0[2 * p + 1] + bias, 0.0f);
        float v10 = fmaxf(acc1[2 * p]     + bias, 0.0f);
        float v11 = fmaxf(acc1[2 * p + 1] + bias, 0.0f);
        float pooled = fmaxf(fmaxf(v00, v01), fmaxf(v10, v11));
        if (mlo < 8) {
            int px = t * 8 + p + 4 * hi;
            h2[(((size_t)img * 64 + r) * 64 + px) * 8 + mlo] = pooled;
        }
    }
}

// ---------------------------------------------------------------------------
// Kernel 4: conv 1x1 (8->4) + bias -> "low" output (NCHW flat, NT stores).
// ---------------------------------------------------------------------------
__global__ __launch_bounds__(256)
void k_conv3(const float* __restrict__ h2, const float* __restrict__ w3,
             const float* __restrict__ b3, float* __restrict__ low) {
    int idx = blockIdx.x * blockDim.x + threadIdx.x;   // 96*4096
    int p   = idx & 4095;
    int img = idx >> 12;
    const float* in = h2 + (size_t)idx * 8;
    v4f f0 = *(const v4f*)(in);
    v4f f1 = *(const v4f*)(in + 4);
    float v[8] = { f0[0], f0[1], f0[2], f0[3], f1[0], f1[1], f1[2], f1[3] };
    #pragma unroll
    for (int oc = 0; oc < 4; ++oc) {
        float s = b3[oc];
        #pragma unroll
        for (int ic = 0; ic < 8; ++ic) s += v[ic] * w3[oc * 8 + ic];
        __builtin_nontemporal_store(s, &low[((size_t)img * 4 + oc) * 4096 + p]);
    }
}

// ---------------------------------------------------------------------------
extern "C" void kernel_launch(void* const* d_in, const int* in_sizes, int n_in,
                              void* d_out, int out_size, void* d_ws, size_t ws_size,
                              hipStream_t stream) {
    const float* x  = (const float*)d_in[0];
    const float* w1 = (const float*)d_in[1];
    const float* b1 = (const float*)d_in[2];
    const float* w2 = (const float*)d_in[3];
    const float* b2 = (const float*)d_in[4];
    const float* w3 = (const float*)d_in[5];
    const float* b3 = (const float*)d_in[6];

    float* low  = (float*)d_out;                 // 32*49152 floats
    float* high = low + 1572864;

    float* ws  = (float*)d_ws;
    float* cA  = ws + WS_CA;
    float* h1  = ws + WS_H1;
    float* h2  = ws + WS_H2;
    float* wB2 = ws + WS_WB2;
    float* wB1 = ws + WS_WB1;

    k_zero_ca<<<  386, 256, 0, stream>>>(cA);
    k_zero_h1<<< 3096, 256, 0, stream>>>(h1);
    k_prepw  <<<   10, 256, 0, stream>>>(w1, w2, wB1, wB2);
    k_dwt    <<<24576, 256, 0, stream>>>(x, cA, high);
    k_conv1  <<<24576, 256, 0, stream>>>(cA, wB1, b1, h1);
    k_conv2  <<< 6144, 256, 0, stream>>>(h1, wB2, b2, h2);
    k_conv3  <<< 1536, 256, 0, stream>>>(h2, w3, b3, low);
}